// PeakExtractor_74191265071636
// MI455X (gfx1250) — compile-verified
//
#include <hip/hip_runtime.h>
#include <hip/hip_bf16.h>
#include <stdint.h>

// Peak extraction: one workgroup per (b,n,c) channel.
// TDM (tensor_load_to_lds) streams the channel's 128KB density tile into LDS.
// Argmax is hierarchical: 1024 segment keys (32 elems each, strided layout for
// conflict-free ds_load_b128) + per-iteration incremental repair of only the
// segments touched by the 7x7x7 Chebyshev suppression box.

typedef __attribute__((ext_vector_type(4))) uint32_t u32x4;
typedef __attribute__((ext_vector_type(8))) uint32_t u32x8;

#define PK_NEG    (-1e9f)
#define PK_STOP   (-1e8f)
#define PK_TOPK   4
#define PK_MAXG   32768
#define PK_NT     1024            // 32 waves (wave32)
#define PK_CHUNK  4096            // elements per chunk (PK_NT * 4)

// Monotone map f32 -> u32 (order-preserving).
__device__ __forceinline__ uint32_t f32_ord(float f) {
    uint32_t u = __float_as_uint(f);
    uint32_t m = (uint32_t)((int32_t)u >> 31) | 0x80000000u;
    return u ^ m;
}
__device__ __forceinline__ float ord_f32(uint32_t x) {   // exact inverse
    uint32_t u = (x & 0x80000000u) ? (x ^ 0x80000000u) : ~x;
    return __uint_as_float(u);
}
__device__ __forceinline__ unsigned long long umax64(unsigned long long a,
                                                     unsigned long long b) {
    return a > b ? a : b;
}
__device__ __forceinline__ unsigned long long elem_key(float f, int g) {
    return ((unsigned long long)f32_ord(f) << 32) | (uint32_t)(~g);
}

__global__ __launch_bounds__(PK_NT)
void peak_extract_kernel(const float* __restrict__ density,
                         const float* __restrict__ grid_xyz,
                         const unsigned char* __restrict__ sphere_mask,
                         const int* __restrict__ pnx,
                         const int* __restrict__ pny,
                         const int* __restrict__ pnz,
                         float* __restrict__ out_xyz,
                         float* __restrict__ out_score,
                         float* __restrict__ out_mask,
                         int G)
{
    __shared__ float buf[PK_MAXG];                   // 128 KB channel tile
    __shared__ unsigned long long segkey[PK_NT];     // 8 KB: 1024 segment maxima
    __shared__ unsigned long long wkey[PK_NT / 32];  // per-wave partials
    __shared__ unsigned long long peakKey;

    const int tid  = threadIdx.x;
    const int lane = tid & 31;
    const int wid  = tid >> 5;
    const int bnc  = blockIdx.x;

    const int nx = pnx[0], ny = pny[0], nz = pnz[0];
    const int nynz = ny * nz;
    const int nchunks = (G + PK_CHUNK - 1) / PK_CHUNK;

    // ---- TDM: DMA this channel's G floats (G*4 bytes) into LDS ----
    if (wid == 0) {
        const uint64_t gaddr =
            (uint64_t)(uintptr_t)(const void*)(density + (size_t)bnc * (size_t)G);
        const uint32_t ldsb = (uint32_t)(uintptr_t)(void*)&buf[0];
        const uint32_t d0   = (uint32_t)G;

        u32x4 g0;
        g0.x = 1u;                                       // count=1 (valid user D#)
        g0.y = ldsb;                                     // lds_addr
        g0.z = (uint32_t)(gaddr & 0xFFFFFFFFu);          // global_addr[31:0]
        g0.w = (uint32_t)(gaddr >> 32) | 0x80000000u;    // global_addr[56:32] | type=2

        u32x8 g1;
        g1.s0 = 2u << 16;                                // data_size=4B
        g1.s1 = (d0 & 0xFFFFu) << 16;                    // tensor_dim0[15:0]
        g1.s2 = (d0 >> 16) | (1u << 16);                 // tensor_dim0[31:16] | tensor_dim1=1
        g1.s3 = (d0 & 0xFFFFu) << 16;                    // tile_dim0 = G
        g1.s4 = 1u;                                      // tile_dim1=1, tile_dim2=0
        g1.s5 = d0;                                      // tensor_dim0_stride[31:0]
        g1.s6 = 0u;
        g1.s7 = 0u;

        asm volatile("tensor_load_to_lds %0, %1"
                     :: "s"(g0), "s"(g1)
                     : "memory");
        __builtin_amdgcn_s_wait_tensorcnt(0);
    }
    __syncthreads();

    // ---- build segment keys (fused sphere-mask application) ----
    // Segment t = { c*4096 + 4t .. 4t+3 : c }: ds_load_b128, lanes cover 512
    // consecutive bytes -> conflict-free.
    {
        unsigned long long key = 0ull;
        for (int c = 0; c < nchunks; ++c) {
            const int gb = c * PK_CHUNK + 4 * tid;
            if (gb + 3 < G) {
                float4 v = *(const float4*)&buf[gb];
                const uint32_t mb = *(const uint32_t*)(sphere_mask + gb);
                float f0 = v.x, f1 = v.y, f2 = v.z, f3 = v.w;
                if (!(mb & 0x000000FFu)) { f0 = PK_NEG; buf[gb + 0] = PK_NEG; }
                if (!(mb & 0x0000FF00u)) { f1 = PK_NEG; buf[gb + 1] = PK_NEG; }
                if (!(mb & 0x00FF0000u)) { f2 = PK_NEG; buf[gb + 2] = PK_NEG; }
                if (!(mb & 0xFF000000u)) { f3 = PK_NEG; buf[gb + 3] = PK_NEG; }
                key = umax64(key, elem_key(f0, gb + 0));
                key = umax64(key, elem_key(f1, gb + 1));
                key = umax64(key, elem_key(f2, gb + 2));
                key = umax64(key, elem_key(f3, gb + 3));
            }
        }
        segkey[tid] = key;
    }
    __syncthreads();

    // ---- TOPK iterations ----
    for (int k = 0; k < PK_TOPK; ++k) {
        // 1) argmax over 1024 segment keys
        unsigned long long key = segkey[tid];
        for (int off = 16; off > 0; off >>= 1)
            key = umax64(key, (unsigned long long)__shfl_xor((unsigned long long)key, off, 32));
        if (lane == 0) wkey[wid] = key;
        __syncthreads();
        if (wid == 0) {
            unsigned long long k2 = wkey[lane];
            for (int off = 16; off > 0; off >>= 1)
                k2 = umax64(k2, (unsigned long long)__shfl_xor((unsigned long long)k2, off, 32));
            if (lane == 0) peakKey = k2;
        }
        __syncthreads();

        const unsigned long long pk = peakKey;
        const int   pidx  = (int)(~(uint32_t)pk);
        const float score = ord_f32((uint32_t)(pk >> 32));   // exact peak value
        const bool  valid = (score >= PK_STOP);

        // peak voxel coords + clipped box
        const int pi = pidx / nynz;
        const int rr = pidx - pi * nynz;
        const int pj = rr / nz;
        const int pz = rr - pj * nz;
        const int x0 = max(pi - 3, 0), x1 = min(pi + 3, nx - 1);
        const int y0 = max(pj - 3, 0), y1 = min(pj + 3, ny - 1);
        const int z0 = max(pz - 3, 0), z1 = min(pz + 3, nz - 1);
        const int nry = y1 - y0 + 1;
        const int nrows = (x1 - x0 + 1) * nry;

        // 2) suppress box elements in buf
        if (valid && tid < 343) {
            const int x = pi + (tid / 49) - 3;
            const int y = pj + ((tid / 7) % 7) - 3;
            const int z = pz + (tid % 7) - 3;
            if (x >= 0 && x < nx && y >= 0 && y < ny && z >= 0 && z < nz)
                buf[x * nynz + y * nz + z] = PK_NEG;
        }

        // 3) outputs (thread 0)
        if (tid == 0) {
            const size_t o = (size_t)bnc * PK_TOPK + k;
            if (valid) {
                out_xyz[o * 3 + 0] = grid_xyz[(size_t)pidx * 3 + 0];
                out_xyz[o * 3 + 1] = grid_xyz[(size_t)pidx * 3 + 1];
                out_xyz[o * 3 + 2] = grid_xyz[(size_t)pidx * 3 + 2];
                out_score[o] = score;
                out_mask[o]  = 1.0f;
            } else {
                out_xyz[o * 3 + 0] = 0.0f;
                out_xyz[o * 3 + 1] = 0.0f;
                out_xyz[o * 3 + 2] = 0.0f;
                out_score[o] = PK_NEG;
                out_mask[o]  = 0.0f;
            }
        }
        __syncthreads();   // suppression visible before segment repair

        // 4) repair only the segments touched by the box.
        //    Row r: 7 contiguous z-elements -> <= 3 aligned 4-element groups.
        if (valid && tid < nrows * 3) {
            const int r = tid / 3, j = tid - 3 * r;
            const int x = x0 + r / nry;
            const int y = y0 + r - (r / nry) * nry;
            const int g0c = x * nynz + y * nz + z0;
            const int g1c = g0c + (z1 - z0);
            const int off = g0c & 3;
            const int ngroups = (off + (g1c - g0c + 1) + 3) >> 2;
            if (j < ngroups) {
                const int gr  = g0c + (j ? 4 * j - off : 0);     // element in group j
                const int seg = (gr & (PK_CHUNK - 1)) >> 2;      // owning segment
                unsigned long long nk = 0ull;
                for (int c = 0; c < nchunks; ++c) {
                    const int gb = c * PK_CHUNK + 4 * seg;
                    if (gb + 3 < G) {
                        float4 v = *(const float4*)&buf[gb];
                        nk = umax64(nk, elem_key(v.x, gb + 0));
                        nk = umax64(nk, elem_key(v.y, gb + 1));
                        nk = umax64(nk, elem_key(v.z, gb + 2));
                        nk = umax64(nk, elem_key(v.w, gb + 3));
                    }
                }
                segkey[seg] = nk;    // duplicates write identical values
            }
        }
        __syncthreads();   // repaired keys visible before next argmax
    }
}

extern "C" void kernel_launch(void* const* d_in, const int* in_sizes, int n_in,
                              void* d_out, int out_size, void* d_ws, size_t ws_size,
                              hipStream_t stream) {
    const float*         density  = (const float*)d_in[0];
    const float*         grid_xyz = (const float*)d_in[1];
    const unsigned char* sphere   = (const unsigned char*)d_in[2];
    const int*           pnx      = (const int*)d_in[3];
    const int*           pny      = (const int*)d_in[4];
    const int*           pnz      = (const int*)d_in[5];

    const int G   = in_sizes[1] / 3;       // grid_xyz is [G,3]
    const int BNC = in_sizes[0] / G;       // density is [B*N*C, G]

    float* out       = (float*)d_out;
    float* out_xyz   = out;                                  // [BNC, K, 3]
    float* out_score = out + (size_t)BNC * PK_TOPK * 3;      // [BNC, K]
    float* out_mask  = out_score + (size_t)BNC * PK_TOPK;    // [BNC, K]

    peak_extract_kernel<<<BNC, PK_NT, 0, stream>>>(
        density, grid_xyz, sphere, pnx, pny, pnz,
        out_xyz, out_score, out_mask, G);
}